// Mixture_Model_16475494547798
// MI455X (gfx1250) — compile-verified
//
#include <hip/hip_runtime.h>
#include <hip/hip_bf16.h>
#include <math.h>

// Problem constants (from reference)
#define NN   4096
#define LL_  32      // latent dim L
#define KK   8
#define DC   64
#define DB   64
#define DD   16
#define CC   8
#define BB   32
#define MM   1000

#define COLS_PER_K   (DC + DB + DD * CC)   // 256
#define ROWS         (BB * MM)             // 32000
#define ROW_TILES    (ROWS / 16)           // 2000

#define LOG2PI_F 1.8378770664093453f

// LDS staging: one chunk = one weight region (max 128 columns x 32 L), padded
// to 34 dwords per column (8B-aligned ds_load_b64, near-conflict-free banks).
#define PAD_COL      34
#define CHUNK_DW     (128 * PAD_COL)       // 4352 dwords per buffer
#define NCHUNKS      (KK * 3)              // 24

typedef __attribute__((ext_vector_type(2))) float v2f;
typedef __attribute__((ext_vector_type(8))) float v8f;
typedef __attribute__((ext_vector_type(4))) unsigned int u32x4;
typedef __attribute__((ext_vector_type(8))) int i32x8;
typedef __attribute__((ext_vector_type(4))) int i32x4;

// Output layout (floats):
// [0]=-elbo [1]=LL [2]=KL_z [3]=KL_s | rik (N*K) | term_1 (M) | term_2 (M) | term_3 (M)
#define OUT_RIK   4
#define OUT_T1    (4 + NN * KK)
#define OUT_T2    (OUT_T1 + MM)
#define OUT_T3    (OUT_T2 + MM)

__global__ void mm_zero_kernel(float* out, int n, float* ws) {
    int i = blockIdx.x * blockDim.x + threadIdx.x;
    if (i < n) out[i] = 0.0f;
    if (i == 0) ws[0] = 0.0f;
}

// Issue one 1-D TDM copy (ndwords f32) global -> LDS with column padding:
// after every 32 dwords stored, skip 2 dwords in LDS (-> 34-dword column stride).
__device__ __forceinline__ void tdm_load_1d(unsigned int lds_byte_addr,
                                            const float* gsrc,
                                            unsigned int ndwords) {
    unsigned long long ga = (unsigned long long)(uintptr_t)gsrc;

    u32x4 g0;
    g0[0] = 1u;                                   // count=1, is_restore=0, gather off
    g0[1] = lds_byte_addr;                        // D#.lds_addr
    g0[2] = (unsigned int)ga;                     // global_addr[31:0]
    g0[3] = (unsigned int)((ga >> 32) & 0x1FFFFFFu) | (2u << 30); // addr[56:32] | type=2

    i32x8 g1;
    // workgroup_mask=0 | data_size=2 (4B) | pad_enable | pad_interval=4 (32dw) | pad_amount=1 (2dw)
    g1[0] = (int)((2u << 16) | (1u << 20) | (4u << 22) | (1u << 25));
    g1[1] = (int)((ndwords & 0xFFFFu) << 16);     // tensor_dim0[15:0] at bits 63:48
    g1[2] = (int)((ndwords >> 16) & 0xFFFFu);     // tensor_dim0[31:16]; tensor_dim1=0
    g1[3] = (int)((ndwords & 0xFFFFu) << 16);     // tile_dim0 at bits 127:112
    g1[4] = 0;                                    // tile_dim1=0, tile_dim2=0 (1-D tile)
    g1[5] = (int)ndwords;                         // tensor_dim0_stride lo32
    g1[6] = 0;                                    // stride hi; dim1_stride lo
    g1[7] = 0;

    i32x4 z4 = {0, 0, 0, 0};
    i32x8 z8 = {0, 0, 0, 0, 0, 0, 0, 0};
    __builtin_amdgcn_tensor_load_to_lds(g0, g1, z4, z4, z8, 0);
}

// Main fused kernel: z-gen + GEMM (f32 WMMA 16x16x4, B from TDM-staged LDS)
// + log-lik epilogues + reductions. One wave per 16-row tile; 4 waves/block
// share double-buffered LDS weight chunks.
__global__ __launch_bounds__(128) void mm_main_kernel(
    const int*   __restrict__ index_p,
    const float* __restrict__ X_c,      // (B, Dc)
    const float* __restrict__ X_b,      // (B, Db)
    const int*   __restrict__ X_d,      // (B, Dd)
    const float* __restrict__ qzm,      // (N, L)
    const float* __restrict__ qlv,      // (N, L)
    const float* __restrict__ qsp,      // (N, K)
    const float* __restrict__ W_g,      // (K, Dc, L)
    const float* __restrict__ b_g,      // (K, Dc)
    const float* __restrict__ lvg,      // (K, Dc)
    const float* __restrict__ W_b,      // (K, Db, L)
    const float* __restrict__ b_b,      // (K, Db)
    const float* __restrict__ W_d,      // (K, Dd, C, L)
    const float* __restrict__ b_d,      // (K, Dd, C)
    const float* __restrict__ eps,      // (B, M, L)
    float* __restrict__ ws,             // ws[0] = LL accumulator
    float* __restrict__ out)
{
    __shared__ __align__(16) float smem[2 * CHUNK_DW];   // 34816 B

    const int start = BB * index_p[0];

    const int lane  = threadIdx.x & 31;
    const int half  = lane >> 4;
    const int mrow  = lane & 15;
    const int warp  = threadIdx.x >> 5;
    const int wid   = blockIdx.x * (blockDim.x >> 5) + warp;
    const int r0    = wid * 16;

    // Chunk source table: region r of component k
    auto issue_chunk = [&](int chunk, int buf) {
        int k = chunk / 3, r = chunk - 3 * k;
        unsigned int base = (unsigned int)(uintptr_t)&smem[buf * CHUNK_DW];
        if (r == 0)      tdm_load_1d(base, W_g + (size_t)k * DC * LL_,      DC * LL_);
        else if (r == 1) tdm_load_1d(base, W_b + (size_t)k * DB * LL_,      DB * LL_);
        else             tdm_load_1d(base, W_d + (size_t)k * DD * CC * LL_, DD * CC * LL_);
    };

    // ---- A tile (z, 16 rows x 32 L), WMMA f32 16x16x4 A layout:
    // lanes 0-15 hold K = {4s,4s+1}, lanes 16-31 hold K = {4s+2,4s+3}.
    const int rA = r0 + mrow;
    const int bA = rA / MM;
    v2f a[8];
    #pragma unroll
    for (int s = 0; s < 8; ++s) {
        const int l0 = 4 * s + 2 * half;
        v2f e = *(const v2f*)(eps + (size_t)rA * LL_ + l0);
        const float* qmr = qzm + (size_t)(start + bA) * LL_ + l0;
        const float* lvr = qlv + (size_t)(start + bA) * LL_ + l0;
        a[s] = (v2f){qmr[0] + __expf(0.5f * lvr[0]) * e.x,
                     qmr[1] + __expf(0.5f * lvr[1]) * e.y};
    }

    // C layout: VGPR i <-> row r0 + i + 8*half, col = lane&15 within tile
    int cb[8], cm[8];
    #pragma unroll
    for (int i = 0; i < 8; ++i) {
        int r = r0 + 8 * half + i;
        cb[i] = r / MM;
        cm[i] = r - cb[i] * MM;
    }
    const int colN = lane & 15;

    float t1l[8], t2l[8], t3l[8], LLl[8];
    #pragma unroll
    for (int i = 0; i < 8; ++i) { t1l[i] = 0.f; t2l[i] = 0.f; t3l[i] = 0.f; LLl[i] = 0.f; }

    // Prologue: wave 0 prefetches chunk 0
    if (warp == 0) issue_chunk(0, 0);

    for (int k = 0; k < KK; ++k) {
        float p1[8], p2[8], p3[8];
        #pragma unroll
        for (int i = 0; i < 8; ++i) { p1[i] = 0.f; p2[i] = 0.f; p3[i] = 0.f; }

        for (int reg = 0; reg < 3; ++reg) {
            const int chunk = k * 3 + reg;
            const int buf   = chunk & 1;

            if (warp == 0) {
                if (chunk + 1 < NCHUNKS) {
                    issue_chunk(chunk + 1, (chunk + 1) & 1);
                    __builtin_amdgcn_s_wait_tensorcnt(1);  // current chunk landed
                } else {
                    __builtin_amdgcn_s_wait_tensorcnt(0);
                }
            }
            __syncthreads();   // LDS chunk visible to all 4 waves

            const int ntiles = (reg == 2) ? 8 : 4;
            const int ldsbuf = buf * CHUNK_DW;

            for (int ct = 0; ct < ntiles; ++ct) {
                const int colr = ct * 16 + colN;   // column within region

                // B tile from LDS (f32 4x16 per step: lanes 0-15 K={4s,4s+1},
                // lanes 16-31 K={4s+2,4s+3}); 34-dword column stride.
                const float* cbase = &smem[ldsbuf + colr * PAD_COL + 2 * half];
                v2f bt[8];
                #pragma unroll
                for (int s = 0; s < 8; ++s)
                    bt[s] = *(const v2f*)(cbase + 4 * s);

                // 8 chained f32 WMMAs over L = 32
                v8f acc = {};
                #pragma unroll
                for (int s = 0; s < 8; ++s) {
                    acc = __builtin_amdgcn_wmma_f32_16x16x4_f32(
                        false, a[s], false, bt[s], (short)0, acc, false, false);
                }

                // ---- fused epilogue (region type is uniform)
                if (reg == 0) {
                    // Gaussian
                    const int d = colr;
                    const float bias = b_g[k * DC + d];
                    const float lv   = lvg[k * DC + d];
                    const float iv   = __expf(-lv);
                    #pragma unroll
                    for (int i = 0; i < 8; ++i) {
                        float xc   = X_c[cb[i] * DC + d];
                        float df   = xc - (acc[i] + bias);
                        p1[i] += -0.5f * (LOG2PI_F + lv + df * df * iv);
                    }
                } else if (reg == 1) {
                    // Bernoulli
                    const int d = colr;
                    const float bias = b_b[k * DB + d];
                    #pragma unroll
                    for (int i = 0; i < 8; ++i) {
                        float xb = X_b[cb[i] * DB + d];
                        float lg = acc[i] + bias;
                        float sp = fmaxf(lg, 0.f) + log1pf(__expf(-fabsf(lg)));
                        p2[i] += xb * lg - sp;
                    }
                } else {
                    // Categorical: log-softmax over C=8 adjacent columns
                    const int d  = colr >> 3;
                    const int ch = colr & 7;
                    const float bias = b_d[(k * DD + d) * CC + ch];
                    #pragma unroll
                    for (int i = 0; i < 8; ++i) {
                        float lg = acc[i] + bias;
                        float mx = lg;
                        mx = fmaxf(mx, __shfl_xor(mx, 1));
                        mx = fmaxf(mx, __shfl_xor(mx, 2));
                        mx = fmaxf(mx, __shfl_xor(mx, 4));
                        float se = __expf(lg - mx);
                        se += __shfl_xor(se, 1);
                        se += __shfl_xor(se, 2);
                        se += __shfl_xor(se, 4);
                        float lse = mx + __logf(se);
                        int xd = X_d[cb[i] * DD + d];
                        p3[i] += (ch == xd) ? (lg - lse) : 0.f;
                    }
                }
            }
            __syncthreads();   // all waves done with this buffer before reuse
        }

        // ---- fold component k
        #pragma unroll
        for (int i = 0; i < 8; ++i) {
            float qsv = qsp[(size_t)(start + cb[i]) * KK + k];
            LLl[i] += qsv * (p1[i] + p2[i] + p3[i]);
            t1l[i] += p1[i];
            t2l[i] += p2[i];
            t3l[i] += p3[i];
        }
    }

    // ---- cross-lane reduce over the 16 columns (each half = one row set)
    #pragma unroll
    for (int mask = 1; mask <= 8; mask <<= 1) {
        #pragma unroll
        for (int i = 0; i < 8; ++i) {
            t1l[i] += __shfl_xor(t1l[i], mask);
            t2l[i] += __shfl_xor(t2l[i], mask);
            t3l[i] += __shfl_xor(t3l[i], mask);
            LLl[i] += __shfl_xor(LLl[i], mask);
        }
    }

    float llsum = 0.f;
    #pragma unroll
    for (int i = 0; i < 8; ++i) llsum += LLl[i];
    llsum += __shfl_xor(llsum, 16);
    if (lane == 0) atomicAdd(ws, llsum);

    if ((lane & 15) == 0) {
        #pragma unroll
        for (int i = 0; i < 8; ++i) {
            int m = cm[i];
            atomicAdd(out + OUT_T1 + m, t1l[i]);
            atomicAdd(out + OUT_T2 + m, t2l[i]);
            atomicAdd(out + OUT_T3 + m, t3l[i]);
        }
    }
}

// Finalize: KL_z, KL_s, rik softmax rows, scalars. One block of B=32 threads.
__global__ void mm_finalize_kernel(
    const int*   __restrict__ index_p,
    const float* __restrict__ qzm,
    const float* __restrict__ qlv,
    const float* __restrict__ qsp,
    const float* __restrict__ pmean,
    const float* __restrict__ pvar,
    const float* __restrict__ pmu,
    const float* __restrict__ ws,
    float* __restrict__ out)
{
    const int start = BB * index_p[0];
    const int b = threadIdx.x;
    __shared__ float s_klz[BB], s_kls[BB];

    float acc = 0.f;
    const size_t r = (size_t)(start + b) * LL_;
    for (int l = 0; l < LL_; ++l) {
        float qm = qzm[r + l], lv = qlv[r + l];
        float mp = pmean[r + l], vp = pvar[r + l];
        float d  = qm - mp;
        acc += __logf(vp) - lv + (__expf(lv) + d * d) / vp - 1.0f;
    }
    s_klz[b] = 0.5f * acc;

    const size_t rk = (size_t)(start + b) * KK;
    float qrow[KK], prow[KK];
    float qsum = 0.f, psum = 0.f, mx = -1e30f;
    for (int k = 0; k < KK; ++k) {
        qrow[k] = qsp[rk + k];
        prow[k] = pmu[rk + k];
        qsum += qrow[k];
        psum += prow[k];
        mx = fmaxf(mx, qrow[k]);
    }
    float se = 0.f;
    for (int k = 0; k < KK; ++k) se += __expf(qrow[k] - mx);
    float kls = 0.f;
    for (int k = 0; k < KK; ++k) {
        float qn = qrow[k] / qsum;
        float pn = prow[k] / psum;
        kls += qn * (__logf(qn) - __logf(pn));
        out[OUT_RIK + rk + k] = __expf(qrow[k] - mx) / se;
    }
    s_kls[b] = kls;

    __syncthreads();
    if (b == 0) {
        float LL = ws[0];
        float sz = 0.f, ss = 0.f;
        for (int i = 0; i < BB; ++i) { sz += s_klz[i]; ss += s_kls[i]; }
        float elbo = LL - sz - ss;
        out[0] = -elbo;
        out[1] = LL;
        out[2] = s_klz[BB - 1];
        out[3] = s_kls[BB - 1];
    }
}

extern "C" void kernel_launch(void* const* d_in, const int* in_sizes, int n_in,
                              void* d_out, int out_size, void* d_ws, size_t ws_size,
                              hipStream_t stream) {
    const int*   index = (const int*)  d_in[0];
    const float* X_c   = (const float*)d_in[1];
    const float* X_b   = (const float*)d_in[2];
    const int*   X_d   = (const int*)  d_in[3];
    const float* qzm   = (const float*)d_in[4];
    const float* qlv   = (const float*)d_in[5];
    const float* qsp   = (const float*)d_in[6];
    const float* pmean = (const float*)d_in[7];
    const float* pvar  = (const float*)d_in[8];
    const float* pmu   = (const float*)d_in[9];
    const float* W_g   = (const float*)d_in[10];
    const float* b_g   = (const float*)d_in[11];
    const float* lvg   = (const float*)d_in[12];
    const float* W_b   = (const float*)d_in[13];
    const float* b_b   = (const float*)d_in[14];
    const float* W_d   = (const float*)d_in[15];
    const float* b_d   = (const float*)d_in[16];
    const float* eps   = (const float*)d_in[17];
    float* out = (float*)d_out;
    float* ws  = (float*)d_ws;

    mm_zero_kernel<<<(out_size + 255) / 256, 256, 0, stream>>>(out, out_size, ws);

    mm_main_kernel<<<ROW_TILES / 4, 128, 0, stream>>>(
        index, X_c, X_b, X_d, qzm, qlv, qsp,
        W_g, b_g, lvg, W_b, b_b, W_d, b_d, eps, ws, out);

    mm_finalize_kernel<<<1, BB, 0, stream>>>(
        index, qzm, qlv, qsp, pmean, pvar, pmu, ws, out);
}